// SAGE_HBP_23055384445770
// MI455X (gfx1250) — compile-verified
//
#include <hip/hip_runtime.h>
#include <math.h>

typedef float v2f __attribute__((ext_vector_type(2)));
typedef float v8f __attribute__((ext_vector_type(8)));

constexpr int Nn   = 50000;
constexpr int Fdim = 256;
constexpr int Hdim = 256;
constexpr int Cdim = 40;
constexpr int Edg  = 800000;
constexpr float kMinNorm = 1e-15f;
constexpr float kMaxNorm = 0.996f;   // (1 - 4e-3) / sqrt(c), c = 1

// ---- CDNA5 async global->LDS copy (ASYNCcnt-tracked), 16B per lane ----
__device__ __forceinline__ void async_ld16(unsigned int lds_off,
                                           unsigned long long gaddr) {
  asm volatile("global_load_async_to_lds_b128 %0, %1, off"
               :: "v"(lds_off), "v"(gaddr)
               : "memory");
}
__device__ __forceinline__ void wait_async0() {
  asm volatile("s_wait_asynccnt 0x0" ::: "memory");
}
__device__ __forceinline__ unsigned int lds_off32(const void* p) {
  // generic pointer to LDS: low 32 bits are the LDS byte address
  return (unsigned int)(size_t)p;
}

// ---------------- utility kernels ----------------

__global__ void zero_kernel(float* __restrict__ p, int count) {
  int i = blockIdx.x * blockDim.x + threadIdx.x;
  int stride = gridDim.x * blockDim.x;
  for (; i < count; i += stride) p[i] = 0.f;
}

__global__ void count_deg_kernel(const long long* __restrict__ dst,
                                 float* __restrict__ cnt) {
  int e = blockIdx.x * blockDim.x + threadIdx.x;
  if (e < Edg) atomicAdd(&cnt[(int)dst[e]], 1.0f);
}

// one thread = (edge, 4-float chunk); F == H == 256 so shared by both layers
__global__ void scatter_kernel(const float* __restrict__ feat,
                               const long long* __restrict__ src,
                               const long long* __restrict__ dst,
                               float* __restrict__ msg) {
  int gid = blockIdx.x * blockDim.x + threadIdx.x;
  int e = gid >> 6;            // Fdim/4 == 64 chunks per edge
  int c = gid & 63;
  if (e >= Edg) return;
  int s = (int)src[e];
  int d = (int)dst[e];
  const float4 v = *(const float4*)(feat + (size_t)s * Fdim + c * 4);
  float* p = msg + (size_t)d * Fdim + c * 4;
  atomicAdd(p + 0, v.x);
  atomicAdd(p + 1, v.y);
  atomicAdd(p + 2, v.z);
  atomicAdd(p + 3, v.w);
}

__global__ void mean_kernel(float* __restrict__ msg, const float* __restrict__ cnt) {
  int i = blockIdx.x * blockDim.x + threadIdx.x;
  if (i >= Nn * Fdim) return;
  int row = i >> 8;            // / 256
  msg[i] = msg[i] / fmaxf(cnt[row], 1.0f);
}

// ---------------- fused dual GEMM (SAGE layer) ----------------
// Cout[M,Co] = act(A1@B1 + A2@B2 + bias), A* are [Nn,256], B* are [256,Co]
// block = 256 thr = 8 waves; block tile 128x64; wave owns 2x2 WMMA tiles.
// A/B tiles staged with async direct-to-LDS copies; OOB handled by address
// clamping (garbage rows/cols only affect never-stored output rows/cols).
__global__ __launch_bounds__(256) void gemm_dual_kernel(
    const float* __restrict__ A1, const float* __restrict__ B1,
    const float* __restrict__ A2, const float* __restrict__ B2,
    const float* __restrict__ bias, float* __restrict__ Cout,
    int Co, int relu)
{
  constexpr int K = 256;
  constexpr int ASTR = 36;   // 128x32 A tile, 16B-aligned rows, conflict-free
  constexpr int BSTR = 68;   // 32x64 B tile
  __shared__ __align__(16) float As[128 * ASTR];
  __shared__ __align__(16) float Bs[32 * BSTR];
  const int tid  = threadIdx.x;
  const int lane = tid & 31;
  const int wave = tid >> 5;
  const int wr   = wave >> 1;       // 0..3 : 32-row slab
  const int wc   = wave & 1;        // 0..1 : 32-col slab
  const int ln   = lane & 15;
  const int hb   = lane >> 4;
  const int rowBase = blockIdx.x * 128;
  const int colBase = blockIdx.y * 64;

  const unsigned int asBase = lds_off32(As);
  const unsigned int bsBase = lds_off32(Bs);

  v8f acc00 = {0.f,0.f,0.f,0.f,0.f,0.f,0.f,0.f};
  v8f acc01 = acc00, acc10 = acc00, acc11 = acc00;

  const long long bMax = (long long)K * Co - 4;   // 16B-aligned clamp target
  const int ar0 = (wr * 32 + ln) * ASTR;
  const int ar1 = ar0 + 16 * ASTR;
  const int bc0 = wc * 32 + ln;

  for (int sel = 0; sel < 2; ++sel) {
    const float* __restrict__ A = sel ? A2 : A1;
    const float* __restrict__ B = sel ? B2 : B1;
    for (int kb = 0; kb < K; kb += 32) {
      __syncthreads();   // previous tile reads complete before overwrite
      // stage A tile 128x32 : 1024 b128 chunks, 4 per thread
      #pragma unroll
      for (int i = 0; i < 4; ++i) {
        int c  = tid + 256 * i;
        int r  = c >> 3;
        int cc = (c & 7) * 4;
        int gr = rowBase + r;
        gr = (gr < Nn) ? gr : (Nn - 1);
        async_ld16(asBase + (unsigned)(r * ASTR + cc) * 4u,
                   (unsigned long long)(size_t)(A + (size_t)gr * K + kb + cc));
      }
      // stage B tile 32x64 : 512 b128 chunks, 2 per thread
      #pragma unroll
      for (int i = 0; i < 2; ++i) {
        int c  = tid + 256 * i;
        int r  = c >> 4;
        int cc = (c & 15) * 4;
        long long idx = (long long)(kb + r) * Co + colBase + cc;
        if (idx > bMax) idx = bMax;
        async_ld16(bsBase + (unsigned)(r * BSTR + cc) * 4u,
                   (unsigned long long)(size_t)(B + idx));
      }
      wait_async0();
      __syncthreads();

      #pragma unroll
      for (int ks = 0; ks < 8; ++ks) {
        int k0 = ks * 4 + 2 * hb;
        v2f a0, a1, b0, b1;
        a0.x = As[ar0 + k0];          a0.y = As[ar0 + k0 + 1];
        a1.x = As[ar1 + k0];          a1.y = As[ar1 + k0 + 1];
        b0.x = Bs[k0 * BSTR + bc0];          b0.y = Bs[(k0 + 1) * BSTR + bc0];
        b1.x = Bs[k0 * BSTR + bc0 + 16];     b1.y = Bs[(k0 + 1) * BSTR + bc0 + 16];
        acc00 = __builtin_amdgcn_wmma_f32_16x16x4_f32(false, a0, false, b0, (short)0, acc00, false, false);
        acc01 = __builtin_amdgcn_wmma_f32_16x16x4_f32(false, a0, false, b1, (short)0, acc01, false, false);
        acc10 = __builtin_amdgcn_wmma_f32_16x16x4_f32(false, a1, false, b0, (short)0, acc10, false, false);
        acc11 = __builtin_amdgcn_wmma_f32_16x16x4_f32(false, a1, false, b1, (short)0, acc11, false, false);
      }
    }
  }

  #pragma unroll
  for (int nc = 0; nc < 2; ++nc) {
    int col = colBase + wc * 32 + nc * 16 + ln;
    if (col < Co) {
      float bv = bias[col];
      #pragma unroll
      for (int mr = 0; mr < 2; ++mr) {
        const v8f& acc = mr ? (nc ? acc11 : acc10) : (nc ? acc01 : acc00);
        #pragma unroll
        for (int v = 0; v < 8; ++v) {
          int row = rowBase + wr * 32 + mr * 16 + v + 8 * hb;
          if (row < Nn) {
            float val = acc[v] + bv;
            if (relu) val = fmaxf(val, 0.f);
            Cout[(size_t)row * Co + col] = val;
          }
        }
      }
    }
  }
}

// ---------------- proj + logmap0 per-row scalar ----------------
// ||h h^T||_F = ||h||^2, so proj/logmap collapse to one scalar per node
__global__ void scale_kernel(const float* __restrict__ h, float* __restrict__ sc) {
  int n = blockIdx.x * blockDim.x + threadIdx.x;
  if (n >= Nn) return;
  const float* hp = h + (size_t)n * Cdim;
  float q = 0.f;
  for (int i = 0; i < Cdim; ++i) { float v = hp[i]; q += v * v; }
  float norm = fmaxf(q, kMinNorm);
  float s1 = (norm > kMaxNorm) ? (kMaxNorm / norm) : 1.0f;
  float pn = fmaxf(s1 * q, kMinNorm);
  float arg = fminf(fmaxf(pn, -1.0f + 1e-6f), 1.0f - 1e-6f);
  sc[n] = s1 * (atanhf(arg) / pn);
}

// ---------------- outer-product GEMM ----------------
// y[n,k] = s_n * sum_p (h[n,p/40]*h[n,p%40]) * W[p,k] + b[k]
// A fragments generated on the fly from h tile in LDS (no 320MB outer matrix)
__global__ __launch_bounds__(256) void outer_gemm_kernel(
    const float* __restrict__ h, const float* __restrict__ sc,
    const float* __restrict__ W, const float* __restrict__ bias,
    float* __restrict__ out)
{
  constexpr int KK = Cdim * Cdim;   // 1600
  constexpr int KC = 320;           // K chunk staged in LDS
  __shared__ __align__(16) float Ws[KC * Cdim];   // 50 KB
  __shared__ float Ht[128 * 41];    // 128 nodes, stride 41 (bank-conflict-free)
  const int tid  = threadIdx.x;
  const int lane = tid & 31;
  const int wave = tid >> 5;        // 8 waves -> 8 groups of 16 nodes
  const int ln   = lane & 15;
  const int hb   = lane >> 4;
  const int nodeBase = blockIdx.x * 128;
  const unsigned int wsBase = lds_off32(Ws);

  for (int idx = tid; idx < 128 * Cdim; idx += 256) {
    int r = idx / Cdim;
    int c = idx - r * Cdim;
    int gn = nodeBase + r;
    Ht[r * 41 + c] = (gn < Nn) ? h[(size_t)gn * Cdim + c] : 0.f;
  }

  v8f acc0 = {0.f,0.f,0.f,0.f,0.f,0.f,0.f,0.f};
  v8f acc1 = acc0, acc2 = acc0;
  const float* hrow = &Ht[(wave * 16 + ln) * 41];

  for (int kb = 0; kb < KK; kb += KC) {
    __syncthreads();
    // stage W chunk: 12800 floats = 3200 b128 chunks (contiguous, in-bounds)
    #pragma unroll
    for (int i = 0; i < 13; ++i) {
      int c = tid + 256 * i;
      if (c < (KC * Cdim) / 4) {
        async_ld16(wsBase + (unsigned)c * 16u,
                   (unsigned long long)(size_t)(W + (size_t)kb * Cdim + c * 4));
      }
    }
    wait_async0();
    __syncthreads();
    #pragma unroll 4
    for (int ks = 0; ks < KC / 4; ++ks) {
      unsigned p0 = (unsigned)(kb + ks * 4 + 2 * hb);
      unsigned p1 = p0 + 1;
      unsigned i0 = p0 / 40u, j0 = p0 - i0 * 40u;
      unsigned i1 = p1 / 40u, j1 = p1 - i1 * 40u;
      v2f a;
      a.x = hrow[i0] * hrow[j0];
      a.y = hrow[i1] * hrow[j1];
      const float* w0 = &Ws[(ks * 4 + 2 * hb) * Cdim];
      const float* w1 = w0 + Cdim;
      v2f b0, b1, b2;
      b0.x = w0[ln];      b0.y = w1[ln];
      b1.x = w0[ln + 16]; b1.y = w1[ln + 16];
      bool inb = (ln + 32) < Cdim;
      b2.x = inb ? w0[ln + 32] : 0.f;
      b2.y = inb ? w1[ln + 32] : 0.f;
      acc0 = __builtin_amdgcn_wmma_f32_16x16x4_f32(false, a, false, b0, (short)0, acc0, false, false);
      acc1 = __builtin_amdgcn_wmma_f32_16x16x4_f32(false, a, false, b1, (short)0, acc1, false, false);
      acc2 = __builtin_amdgcn_wmma_f32_16x16x4_f32(false, a, false, b2, (short)0, acc2, false, false);
    }
  }

  #pragma unroll
  for (int v = 0; v < 8; ++v) {
    int node = nodeBase + wave * 16 + v + 8 * hb;
    if (node < Nn) {
      float s = sc[node];
      float* op = out + (size_t)node * Cdim;
      op[ln]      = s * acc0[v] + bias[ln];
      op[ln + 16] = s * acc1[v] + bias[ln + 16];
      if (ln + 32 < Cdim) op[ln + 32] = s * acc2[v] + bias[ln + 32];
    }
  }
}

// ---------------- expmap0 + proj + log_softmax ----------------
__global__ void epilogue_kernel(float* __restrict__ y) {
  int n = blockIdx.x * blockDim.x + threadIdx.x;
  if (n >= Nn) return;
  float* yp = y + (size_t)n * Cdim;
  float v[Cdim];
  float q = 0.f;
  for (int i = 0; i < Cdim; ++i) { v[i] = yp[i]; q += v[i] * v[i]; }
  float nr = sqrtf(q);
  float un = fmaxf(nr, kMinNorm);
  float t  = tanhf(un);
  float scale = t / un;                       // expmap0 factor
  float nrm = fmaxf(scale * nr, kMinNorm);    // ||expmap0(y)||
  float pr = (nrm > kMaxNorm) ? (kMaxNorm / nrm) : 1.0f;
  float f = scale * pr;
  float mx = -1e30f;
  for (int i = 0; i < Cdim; ++i) { v[i] *= f; mx = fmaxf(mx, v[i]); }
  float sum = 0.f;
  for (int i = 0; i < Cdim; ++i) sum += expf(v[i] - mx);
  float lse = logf(sum);
  for (int i = 0; i < Cdim; ++i) yp[i] = v[i] - mx - lse;
}

// ---------------- launcher ----------------

extern "C" void kernel_launch(void* const* d_in, const int* in_sizes, int n_in,
                              void* d_out, int out_size, void* d_ws, size_t ws_size,
                              hipStream_t stream) {
  (void)in_sizes; (void)n_in; (void)out_size; (void)ws_size;
  const float* x        = (const float*)d_in[0];
  const long long* ei   = (const long long*)d_in[1];   // int64 [2,E]
  const float* Wl0      = (const float*)d_in[2];
  const float* bl0      = (const float*)d_in[3];
  const float* Wr0      = (const float*)d_in[4];
  const float* Wl1      = (const float*)d_in[5];
  const float* bl1      = (const float*)d_in[6];
  const float* Wr1      = (const float*)d_in[7];
  const float* W_lin    = (const float*)d_in[8];
  const float* b_lin    = (const float*)d_in[9];
  const long long* src  = ei;
  const long long* dst  = ei + Edg;

  float* ws   = (float*)d_ws;
  float* bufA = ws;                          // Nn*256  (msg/mean, reused both layers)
  float* h0   = bufA + (size_t)Nn * Fdim;    // Nn*256
  float* h1   = h0 + (size_t)Nn * Hdim;      // Nn*40
  float* cnt  = h1 + (size_t)Nn * Cdim;      // Nn
  float* scv  = cnt + Nn;                    // Nn
  float* y    = (float*)d_out;               // Nn*40

  const int TB = 256;

  // ---- layer 0 ----
  zero_kernel<<<4096, TB, 0, stream>>>(bufA, Nn * Fdim);
  zero_kernel<<<64,   TB, 0, stream>>>(cnt, Nn);
  count_deg_kernel<<<(Edg + TB - 1) / TB, TB, 0, stream>>>(dst, cnt);
  {
    long long tot = (long long)Edg * (Fdim / 4);
    scatter_kernel<<<(int)((tot + TB - 1) / TB), TB, 0, stream>>>(x, src, dst, bufA);
  }
  mean_kernel<<<(Nn * Fdim + TB - 1) / TB, TB, 0, stream>>>(bufA, cnt);
  {
    dim3 grid((Nn + 127) / 128, Hdim / 64);
    gemm_dual_kernel<<<grid, TB, 0, stream>>>(bufA, Wl0, x, Wr0, bl0, h0, Hdim, 1);
  }

  // ---- layer 1 ----
  zero_kernel<<<4096, TB, 0, stream>>>(bufA, Nn * Hdim);
  {
    long long tot = (long long)Edg * (Hdim / 4);
    scatter_kernel<<<(int)((tot + TB - 1) / TB), TB, 0, stream>>>(h0, src, dst, bufA);
  }
  mean_kernel<<<(Nn * Hdim + TB - 1) / TB, TB, 0, stream>>>(bufA, cnt);
  {
    dim3 grid((Nn + 127) / 128, 1);   // 64 cols cover Co=40 (pad cols unused)
    gemm_dual_kernel<<<grid, TB, 0, stream>>>(bufA, Wl1, h0, Wr1, bl1, h1, Cdim, 0);
  }

  // ---- hyperbolic head ----
  scale_kernel<<<(Nn + TB - 1) / TB, TB, 0, stream>>>(h1, scv);
  outer_gemm_kernel<<<(Nn + 127) / 128, TB, 0, stream>>>(h1, scv, W_lin, b_lin, y);
  epilogue_kernel<<<(Nn + TB - 1) / TB, TB, 0, stream>>>(y);
}